// Threshold_738734375660
// MI455X (gfx1250) — compile-verified
//
#include <hip/hip_runtime.h>
#include <stdint.h>

// Canny hysteresis thresholding, bit-exact vs. the JAX reference.
//   k_init  : zero the atomic-max slot
//   k_max   : global max of img (uint atomicMax on non-negative float bits)
//   k_masks : per-row 2048-bit STRONG / WEAK masks (32 u64 words per row)
//   k_hyst  : ONE wave32 walks rows 1..H-2; per row the boolean recurrence
//             x[j] = a[j] | (b[j] & x[j-1]) is solved in O(1) with the
//             carry-propagation adder trick (64-bit add per lane + ballot +
//             scalar 32-bit add across lanes). Row masks are prefetched with
//             gfx1250 async global->LDS (ASYNCcnt) through a DEPTH=8 ring,
//             so the serial wave waits only on its ALU chain, not on L2.
//   k_expand: parallel expansion of final strong masks to f32 output.

#define H 2048
#define W 2048
#define WPR 32     // 64-bit words per row (32 * 64 = 2048 columns)
#define DEPTH 8    // async prefetch ring depth (16 loads in flight)

typedef unsigned long long u64;

__global__ void k_init(unsigned* maxbits) { *maxbits = 0u; }

__global__ void k_max(const float* __restrict__ img, unsigned* __restrict__ maxbits, int n4) {
    int tid = blockIdx.x * blockDim.x + threadIdx.x;
    int stride = gridDim.x * blockDim.x;
    const float4* p = (const float4*)img;
    float m = 0.0f;
    for (int i = tid; i < n4; i += stride) {
        float4 v = p[i];
        m = fmaxf(m, fmaxf(fmaxf(v.x, v.y), fmaxf(v.z, v.w)));
    }
    #pragma unroll
    for (int off = 16; off > 0; off >>= 1)
        m = fmaxf(m, __shfl_xor(m, off, 32));
    if ((threadIdx.x & 31) == 0)
        atomicMax(maxbits, __float_as_uint(m));   // valid: all values >= 0
}

__global__ void k_masks(const float* __restrict__ img,
                        const unsigned* __restrict__ maxbits,
                        u64* __restrict__ S, u64* __restrict__ Wk) {
    int idx = blockIdx.x * blockDim.x + threadIdx.x;   // one u64 word (64 pixels)
    if (idx >= H * WPR) return;
    float mx   = __uint_as_float(*maxbits);
    float high = mx * 0.15f;
    float low  = high * 0.05f;
    const float4* p = (const float4*)(img + (size_t)idx * 64);
    u64 s = 0ull, w = 0ull;
    #pragma unroll
    for (int k = 0; k < 16; ++k) {
        float4 v = p[k];
        float x0 = v.x, x1 = v.y, x2 = v.z, x3 = v.w;
        s |= ((u64)(x0 > high)) << (k * 4 + 0);
        s |= ((u64)(x1 > high)) << (k * 4 + 1);
        s |= ((u64)(x2 > high)) << (k * 4 + 2);
        s |= ((u64)(x3 > high)) << (k * 4 + 3);
        w |= ((u64)(x0 >= low && x0 <= high)) << (k * 4 + 0);
        w |= ((u64)(x1 >= low && x1 <= high)) << (k * 4 + 1);
        w |= ((u64)(x2 >= low && x2 <= high)) << (k * 4 + 2);
        w |= ((u64)(x3 >= low && x3 <= high)) << (k * 4 + 3);
    }
    S[idx] = s;
    Wk[idx] = w;
}

// Issue one async global->LDS b64 load (gfx1250, tracked by ASYNCcnt).
__device__ __forceinline__ void async_load_b64(void* lds_ptr, const void* gptr) {
    unsigned lds_off = (unsigned)(size_t)lds_ptr;          // low 32 bits = LDS offset
    u64 gaddr = (u64)gptr;
    asm volatile("global_load_async_to_lds_b64 %0, %1, off"
                 :: "v"(lds_off), "v"(gaddr) : "memory");
}

// Sequential hysteresis: single wave32, lane l owns columns [64l, 64l+64).
__global__ void __launch_bounds__(32) k_hyst(const u64* __restrict__ S,
                                             const u64* __restrict__ Wk,
                                             u64* __restrict__ FS) {
    __shared__ u64 bufS[DEPTH][32];
    __shared__ u64 bufW[DEPTH][32];
    const int lane = threadIdx.x;

    // full-row bit shifts: bit c of mask = column c
    auto shl1 = [&](u64 v) -> u64 {          // result[c] = v[c+1]
        unsigned t = __shfl_down((unsigned)(v & 1ull), 1, 32);
        t = (lane == 31) ? 0u : t;
        return (v >> 1) | ((u64)t << 63);
    };
    auto shr1 = [&](u64 v) -> u64 {          // result[c] = v[c-1]
        unsigned t = __shfl_up((unsigned)(v >> 63), 1, 32);
        t = (lane == 0) ? 0u : t;
        return (v << 1) | (u64)t;
    };

    u64 ps = S[0 * WPR + lane];              // updated row above (row 0 = original)
    u64 cs = S[1 * WPR + lane];
    u64 wk = Wk[1 * WPR + lane];
    FS[0 * WPR + lane]       = ps;           // first/last rows pass through
    FS[(size_t)(H - 1) * WPR + lane] = S[(size_t)(H - 1) * WPR + lane];

    u64 interior = ~0ull;
    if (lane == 0)  interior &= ~1ull;
    if (lane == 31) interior &= ~(1ull << 63);

    // prime: rows 2 .. DEPTH+1 into slots 0 .. DEPTH-1 (16 async loads in flight)
    #pragma unroll
    for (int r = 2; r < 2 + DEPTH; ++r) {
        int slot = r - 2;
        async_load_b64(&bufS[slot][lane], &S[(size_t)r * WPR + lane]);
        async_load_b64(&bufW[slot][lane], &Wk[(size_t)r * WPR + lane]);
    }

    for (int i = 1; i <= H - 2; ++i) {
        // row i+1 lives in slot (i-1) % DEPTH; its 2 loads are the oldest pair.
        // Async loads complete in order -> waiting ASYNCcnt <= 2*(DEPTH-1)=14
        // guarantees that pair has landed while 14 younger loads stay in flight.
        asm volatile("s_wait_asynccnt 0xe" ::: "memory");
        int slot = (i - 1) & (DEPTH - 1);
        u64 ns  = bufS[slot][lane];          // original strong mask, row i+1
        u64 wkn = bufW[slot][lane];          // weak mask, row i+1 (for next iter)
        asm volatile("s_wait_dscnt 0x0" ::: "memory");

        // refill the slot just consumed with row i+1+DEPTH (clamped; the
        // clamped duplicates at the tail are never read again before exit)
        {
            int nr = i + 1 + DEPTH; if (nr > H - 1) nr = H - 1;
            async_load_b64(&bufS[slot][lane], &S[(size_t)nr * WPR + lane]);
            async_load_b64(&bufW[slot][lane], &Wk[(size_t)nr * WPR + lane]);
        }

        // strong among the 7 non-left neighbors
        u64 base = shr1(ps) | ps | shl1(ps) | shl1(cs) | shr1(ns) | ns | shl1(ns);
        u64 wki  = wk & interior;
        u64 a    = cs | (wki & base);        // generate
        u64 X    = a | wki;                  // propagate (OR-form)
        u64 Y    = a;

        // per-lane carries of X+Y (64 columns resolved by one add)
        u64 s1 = X + Y;
        unsigned G = (unsigned)(s1 < X);     // chunk generates carry (cin=0)
        unsigned P = (unsigned)(X == ~0ull); // chunk propagates carry
        // cross-lane chunk scan: same adder trick on 32-bit ballot masks (scalar)
        unsigned g32 = (unsigned)__ballot(G);
        unsigned p32 = (unsigned)__ballot(P);
        unsigned Xs = g32 | p32, Ys = g32;
        unsigned cvec32 = (Xs + Ys) ^ Xs ^ Ys;         // bit l = carry into lane l
        u64 cin = (u64)((cvec32 >> lane) & 1u);

        u64 s2   = s1 + cin;
        u64 cout = (u64)((s1 < X) | (s2 < s1));
        u64 cv   = s2 ^ X ^ Y;                         // bit k = carry into bit k
        u64 x    = (cv >> 1) | (cout << 63);           // bit k = carry out of bit k

        u64 psn = cs | (wki & x);            // updated strong mask of row i
        FS[(size_t)i * WPR + lane] = psn;

        ps = psn; cs = ns; wk = wkn;
    }
}

__global__ void k_expand(const u64* __restrict__ FS,
                         const u64* __restrict__ Wk,
                         float* __restrict__ out) {
    int idx = blockIdx.x * blockDim.x + threadIdx.x;   // one u64 word (64 pixels)
    if (idx >= H * WPR) return;
    int r = idx >> 5;        // / WPR
    int w = idx & 31;
    u64 fs = FS[idx];
    u64 wk = Wk[idx];
    bool rowb = (r == 0) | (r == H - 1);
    float4* o = (float4*)(out + (size_t)idx * 64);
    #pragma unroll
    for (int k = 0; k < 16; ++k) {
        float4 v;
        float* vp = &v.x;
        #pragma unroll
        for (int j = 0; j < 4; ++j) {
            int b = k * 4 + j;
            int c = (w << 6) + b;
            bool f  = (fs >> b) & 1ull;
            bool wb = (wk >> b) & 1ull;
            bool bd = rowb | (c == 0) | (c == W - 1);
            vp[j] = f ? 255.0f : ((wb && bd) ? 25.0f : 0.0f);
        }
        o[k] = v;
    }
}

extern "C" void kernel_launch(void* const* d_in, const int* in_sizes, int n_in,
                              void* d_out, int out_size, void* d_ws, size_t ws_size,
                              hipStream_t stream) {
    const float* img = (const float*)d_in[0];
    float* out = (float*)d_out;

    // workspace layout: [0,256) header (atomic max), then S, Wk, FS masks
    char* ws = (char*)d_ws;
    unsigned* maxbits = (unsigned*)ws;
    u64* S  = (u64*)(ws + 256);
    u64* Wk = (u64*)(ws + 256 + (size_t)H * WPR * 8);
    u64* FS = (u64*)(ws + 256 + 2 * (size_t)H * WPR * 8);

    int n4 = in_sizes[0] / 4;
    int words = H * WPR;

    hipLaunchKernelGGL(k_init,   dim3(1),                   dim3(1),   0, stream, maxbits);
    hipLaunchKernelGGL(k_max,    dim3(1024),                dim3(256), 0, stream, img, maxbits, n4);
    hipLaunchKernelGGL(k_masks,  dim3((words + 255) / 256), dim3(256), 0, stream, img, maxbits, S, Wk);
    hipLaunchKernelGGL(k_hyst,   dim3(1),                   dim3(32),  0, stream, S, Wk, FS);
    hipLaunchKernelGGL(k_expand, dim3((words + 255) / 256), dim3(256), 0, stream, FS, Wk, out);
}